// NerfHead_57690000719918
// MI455X (gfx1250) — compile-verified
//
#include <hip/hip_runtime.h>

#define TPB 256
#define ITEMS 16
#define EPB (TPB * ITEMS)   // 4096 elements per block
#define K2_ITEMS 32         // kernel-2 capacity: 256*32 = 8192 block aggregates
#define NWAVES (TPB / 32)   // wave32: 8 waves per 256-thread block

typedef __attribute__((ext_vector_type(4))) float f32x4;
typedef __attribute__((ext_vector_type(4))) int   i32x4;

struct Pair { float v; unsigned f; };

// segmented-product combine (a = earlier, b = later)
__device__ __forceinline__ Pair seg_combine(Pair a, Pair b) {
  Pair r;
  r.v = b.f ? b.v : a.v * b.v;
  r.f = a.f | b.f;
  return r;
}

// wave32 shuffle-up via DS_BPERMUTE_B32 (lane-permute path, no LDS memory)
__device__ __forceinline__ float shfl_up_f(float x, int delta, int lane) {
  int src = lane - delta; src = src < 0 ? lane : src;
  return __int_as_float(__builtin_amdgcn_ds_bpermute(src << 2, __float_as_int(x)));
}
__device__ __forceinline__ unsigned shfl_up_u(unsigned x, int delta, int lane) {
  int src = lane - delta; src = src < 0 ? lane : src;
  return (unsigned)__builtin_amdgcn_ds_bpermute(src << 2, (int)x);
}

// Block-wide segmented EXCLUSIVE scan of per-thread aggregates.
// Returns the exclusive prefix pair for this thread (combine of all earlier threads).
__device__ __forceinline__ Pair block_exclusive_scan(Pair agg, float* sV, unsigned* sF) {
  const int tid  = threadIdx.x;
  const int lane = tid & 31;
  const int wave = tid >> 5;

  Pair inc = agg;                       // wave32 inclusive Hillis-Steele scan
  #pragma unroll
  for (int d = 1; d < 32; d <<= 1) {
    float    pv = shfl_up_f(inc.v, d, lane);
    unsigned pf = shfl_up_u(inc.f, d, lane);
    if (lane >= d) {
      inc.v = inc.f ? inc.v : pv * inc.v;   // combine(prev, cur)
      inc.f |= pf;
    }
  }
  if (lane == 31) { sV[wave] = inc.v; sF[wave] = inc.f; }
  __syncthreads();
  if (tid == 0) {                       // serial exclusive scan over 8 wave aggregates
    Pair p; p.v = 1.0f; p.f = 0u;
    #pragma unroll
    for (int w = 0; w < NWAVES; w++) {
      Pair a; a.v = sV[w]; a.f = sF[w];
      sV[w] = p.v; sF[w] = p.f;
      p = seg_combine(p, a);
    }
  }
  __syncthreads();
  Pair wex; wex.v = sV[wave]; wex.f = sF[wave];
  Pair lex;                             // lane-exclusive within wave
  lex.v = shfl_up_f(inc.v, 1, lane);
  lex.f = shfl_up_u(inc.f, 1, lane);
  if (lane == 0) { lex.v = 1.0f; lex.f = 0u; }
  return seg_combine(wex, lex);
}

// ---------------- Pass 1: per-block segmented aggregate ----------------
// RT-hint loads: installs the 128MB input into the 192MB L2 for pass 3's re-read.
__global__ void __launch_bounds__(TPB)
k1_block_agg(const float* __restrict__ alpha, const int* __restrict__ ray_id,
             int n, float* __restrict__ aggV, unsigned* __restrict__ aggF) {
  __shared__ float sV[NWAVES];
  __shared__ unsigned sF[NWAVES];
  const int tid  = threadIdx.x;
  const int lane = tid & 31;
  const int wave = tid >> 5;
  const int g0   = blockIdx.x * EPB + tid * ITEMS;

  Pair t; t.v = 1.0f; t.f = 0u;
  int prev = 0;
  if (g0 > 0 && g0 < n) prev = ray_id[g0 - 1];

  if (g0 + ITEMS <= n) {
    float a[ITEMS]; int r[ITEMS];
    const f32x4* a4 = reinterpret_cast<const f32x4*>(alpha  + g0);
    const i32x4* r4 = reinterpret_cast<const i32x4*>(ray_id + g0);
    #pragma unroll
    for (int q = 0; q < ITEMS / 4; q++) {
      f32x4 fa = a4[q]; i32x4 ri = r4[q];
      #pragma unroll
      for (int e = 0; e < 4; e++) { a[4*q+e] = fa[e]; r[4*q+e] = ri[e]; }
    }
    #pragma unroll
    for (int j = 0; j < ITEMS; j++) {
      float v = 1.0f - a[j] + 1e-11f;
      int rp = (j == 0) ? prev : r[j-1];
      unsigned f = (((g0 + j) == 0) || (r[j] != rp)) ? 1u : 0u;
      t.v = f ? v : t.v * v;
      t.f |= f;
    }
  } else {
    int rp = prev;
    for (int j = 0; j < ITEMS; j++) {
      int g = g0 + j;
      if (g < n) {
        float v = 1.0f - alpha[g] + 1e-11f;
        int rc = ray_id[g];
        unsigned f = ((g == 0) || (rc != rp)) ? 1u : 0u;
        t.v = f ? v : t.v * v;
        t.f |= f;
        rp = rc;
      }
    }
  }

  // wave inclusive scan, then fold the 8 wave aggregates in order
  #pragma unroll
  for (int d = 1; d < 32; d <<= 1) {
    float    pv = shfl_up_f(t.v, d, lane);
    unsigned pf = shfl_up_u(t.f, d, lane);
    if (lane >= d) { t.v = t.f ? t.v : pv * t.v; t.f |= pf; }
  }
  if (lane == 31) { sV[wave] = t.v; sF[wave] = t.f; }
  __syncthreads();
  if (tid == 0) {
    Pair p; p.v = sV[0]; p.f = sF[0];
    #pragma unroll
    for (int w = 1; w < NWAVES; w++) {
      Pair a; a.v = sV[w]; a.f = sF[w];
      p = seg_combine(p, a);
    }
    aggV[blockIdx.x] = p.v;
    aggF[blockIdx.x] = p.f;
  }
}

// ---------------- Pass 2: exclusive scan of block aggregates (1 block) ----------------
__global__ void __launch_bounds__(TPB)
k2_scan_aggs(const float* __restrict__ aggV, const unsigned* __restrict__ aggF,
             int nb, float* __restrict__ prefV) {
  __shared__ float sV[NWAVES];
  __shared__ unsigned sF[NWAVES];
  const int tid  = threadIdx.x;
  const int base = tid * K2_ITEMS;

  float v[K2_ITEMS];
  unsigned fm = 0u;
  Pair t; t.v = 1.0f; t.f = 0u;
  #pragma unroll
  for (int j = 0; j < K2_ITEMS; j++) {
    int i = base + j;
    float vv = 1.0f; unsigned ff = 0u;
    if (i < nb) { vv = aggV[i]; ff = aggF[i] & 1u; }
    v[j] = vv;
    fm |= ff << j;
    t.v = ff ? vv : t.v * vv;
    t.f |= ff;
  }
  Pair ex = block_exclusive_scan(t, sV, sF);
  float P = ex.v;                       // exclusive-prefix value (leftmost flag never read)
  #pragma unroll
  for (int j = 0; j < K2_ITEMS; j++) {
    int i = base + j;
    if (i < nb) prefV[i] = P;
    unsigned ff = (fm >> j) & 1u;
    P = ff ? v[j] : P * v[j];
  }
}

// ---------------- default alphainv_last = cumprod[0] (last_idx defaults to 0) ----------
__global__ void k_fill_last(const float* __restrict__ alpha,
                            float* __restrict__ lastOut, int nray) {
  int i = blockIdx.x * blockDim.x + threadIdx.x;
  if (i < nray) lastOut[i] = 1.0f - alpha[0] + 1e-11f;
}

// ---------------- Pass 3: apply prefixes, write weights + segment-last gathers --------
// NT-hint loads: inputs are dead after this pass; don't extend their L2 retention
// while the 64MB non-temporal weight stream flows out.
__global__ void __launch_bounds__(TPB)
k3_apply(const float* __restrict__ alpha, const int* __restrict__ ray_id,
         int n, const float* __restrict__ prefV,
         float* __restrict__ weights, float* __restrict__ lastOut) {
  __shared__ float sV[NWAVES];
  __shared__ unsigned sF[NWAVES];
  const int tid  = threadIdx.x;
  const int g0   = blockIdx.x * EPB + tid * ITEMS;
  const bool full = (g0 + ITEMS <= n);

  // global_prefetch: the block prefix is consumed only after the whole local
  // scan; overlap its fetch latency with that work.
  __builtin_prefetch(prefV + blockIdx.x, 0, 3);

  float a[ITEMS];
  unsigned fmask = 0u, lmask = 0u;
  Pair t; t.v = 1.0f; t.f = 0u;
  int prev = 0;
  if (g0 > 0 && g0 < n) prev = ray_id[g0 - 1];

  if (full) {
    int r[ITEMS];
    const f32x4* a4 = reinterpret_cast<const f32x4*>(alpha  + g0);
    const i32x4* r4 = reinterpret_cast<const i32x4*>(ray_id + g0);
    #pragma unroll
    for (int q = 0; q < ITEMS / 4; q++) {
      f32x4 fa = __builtin_nontemporal_load(a4 + q);
      i32x4 ri = __builtin_nontemporal_load(r4 + q);
      #pragma unroll
      for (int e = 0; e < 4; e++) { a[4*q+e] = fa[e]; r[4*q+e] = ri[e]; }
    }
    int nxt = r[ITEMS - 1];
    if (g0 + ITEMS < n) nxt = ray_id[g0 + ITEMS];
    #pragma unroll
    for (int j = 0; j < ITEMS; j++) {
      float v = 1.0f - a[j] + 1e-11f;
      int rp = (j == 0) ? prev : r[j-1];
      int rn = (j == ITEMS - 1) ? nxt : r[j+1];
      unsigned f      = (((g0 + j) == 0)     || (r[j] != rp)) ? 1u : 0u;
      unsigned isLast = (((g0 + j) == n - 1) || (rn   != r[j])) ? 1u : 0u;
      fmask |= f << j;
      lmask |= isLast << j;
      t.v = f ? v : t.v * v;
      t.f |= f;
    }
  } else {
    int rp = prev;
    for (int j = 0; j < ITEMS; j++) {
      int g = g0 + j;
      float av = 0.0f;
      if (g < n) {
        av = alpha[g];
        int rc = ray_id[g];
        float v = 1.0f - av + 1e-11f;
        unsigned f = ((g == 0) || (rc != rp)) ? 1u : 0u;
        unsigned isLast = (g == n - 1) ? 1u : ((ray_id[g + 1] != rc) ? 1u : 0u);
        fmask |= f << j;
        lmask |= isLast << j;
        t.v = f ? v : t.v * v;
        t.f |= f;
        rp = rc;
      }
      a[j] = av;
    }
  }

  Pair ex = block_exclusive_scan(t, sV, sF);
  float bp = prefV[blockIdx.x];
  float P  = ex.f ? ex.v : bp * ex.v;   // == raw inclusive cumprod at (g0-1); 1.0 at g0==0

  float w[ITEMS];
  #pragma unroll
  for (int j = 0; j < ITEMS; j++) {
    float v = 1.0f - a[j] + 1e-11f;
    unsigned f = (fmask >> j) & 1u;
    w[j] = a[j] * P;                    // transmittance = global shift of cumprod (ref-faithful)
    float cum = f ? v : P * v;          // segmented inclusive cumprod
    if ((lmask >> j) & 1u) {
      int g = g0 + j;
      if (g < n) lastOut[ray_id[g]] = cum;   // unique writer per ray; ray_id re-read hits L2
    }
    P = cum;
  }

  // streaming (non-temporal) b128 stores: don't evict the L2-resident inputs
  if (full) {
    f32x4* w4 = reinterpret_cast<f32x4*>(weights + g0);
    #pragma unroll
    for (int q = 0; q < ITEMS / 4; q++) {
      f32x4 pk;
      #pragma unroll
      for (int e = 0; e < 4; e++) pk[e] = w[4*q+e];
      __builtin_nontemporal_store(pk, w4 + q);
    }
  } else {
    for (int j = 0; j < ITEMS; j++) {
      int g = g0 + j;
      if (g < n) weights[g] = w[j];
    }
  }
}

extern "C" void kernel_launch(void* const* d_in, const int* in_sizes, int n_in,
                              void* d_out, int out_size, void* d_ws, size_t ws_size,
                              hipStream_t stream) {
  const float* alpha  = (const float*)d_in[0];
  const int*   ray_id = (const int*)d_in[1];
  const int n    = in_sizes[0];
  const int nray = out_size - n;

  float* weights = (float*)d_out;
  float* lastOut = (float*)d_out + n;

  const int nb = (n + EPB - 1) / EPB;   // 4096 blocks for N=16.7M (k2 capacity 8192)

  float*    aggV  = (float*)d_ws;
  unsigned* aggF  = (unsigned*)(aggV + nb);
  float*    prefV = (float*)(aggF + nb);

  k1_block_agg<<<nb, TPB, 0, stream>>>(alpha, ray_id, n, aggV, aggF);
  k2_scan_aggs<<<1, TPB, 0, stream>>>(aggV, aggF, nb, prefV);
  k_fill_last<<<(nray + 255) / 256, 256, 0, stream>>>(alpha, lastOut, nray);
  k3_apply<<<nb, TPB, 0, stream>>>(alpha, ray_id, n, prefV, weights, lastOut);
}